// PoolerYOLO_38792144617926
// MI455X (gfx1250) — compile-verified
//
#include <hip/hip_runtime.h>
#include <stdint.h>

#define OUTS      7
#define NBIN      49          // 7*7
#define NSAMP     196         // 7*7*2*2
#define CT        32          // channels per block tile
#define LVL2_HW   400         // 20*20
#define BLK       256

// Per-(channel,bin) accumulation. Inlined at two call sites so clang's
// address-space inference emits ds_load_* for the LDS-staged level-2 path
// and global_load_* for the direct path.
__device__ __forceinline__ void accum_item(
    int idx, int roi, int c0, int C, int plane_stride,
    const float* __restrict__ srcbase,
    const int4* __restrict__ s_off, const float4* __restrict__ s_w,
    float* __restrict__ out)
{
    const int cl  = idx / NBIN;
    const int bin = idx - cl * NBIN;
    const float* __restrict__ src = srcbase + (size_t)cl * plane_stride;
    float acc = 0.f;
#pragma unroll
    for (int k = 0; k < 4; ++k) {
        const int4   o = s_off[bin * 4 + k];
        const float4 w = s_w[bin * 4 + k];
        acc += w.x * src[o.x] + w.y * src[o.y] +
               w.z * src[o.z] + w.w * src[o.w];
    }
    out[((size_t)roi * C + (c0 + cl)) * NBIN + bin] = acc;
}

template <int NT>
__global__ __launch_bounds__(BLK) void roi_align_cdna5_kernel(
    const float* __restrict__ feat0, const float* __restrict__ feat1,
    const float* __restrict__ feat2, const float* __restrict__ boxes,
    const int* __restrict__ batch_ids, const int* __restrict__ feat_idx,
    float* __restrict__ out, int C)
{
    __shared__ int4   s_off[NSAMP];              // 4 corner offsets per sample
    __shared__ float4 s_w[NSAMP];                // 4 corner weights (x0.25, masked)
    __shared__ float  s_planes[CT * LVL2_HW];    // level-2: 32 channel planes (51.2 KB)

    const int roi   = blockIdx.x / NT;           // NT is a power of two -> shift
    const int ctile = blockIdx.x - roi * NT;
    const int c0    = ctile << 5;
    const int tid   = threadIdx.x;

    // ---- per-roi uniform scalars ----
    const int fi  = feat_idx[roi];
    const int lvl = (fi < 6399) ? 2 : ((fi < 7999) ? 1 : 0);
    const int b   = batch_ids[roi];

    int H, W; float scale; const float* feat;
    if (lvl == 0)      { H = 80; W = 80; scale = 8.f;  feat = feat0; }
    else if (lvl == 1) { H = 40; W = 40; scale = 16.f; feat = feat1; }
    else               { H = 20; W = 20; scale = 32.f; feat = feat2; }
    const int HW = H * W;

    // ---- level-2: async DMA of 32 full channel planes into LDS (ASYNCcnt),
    //      overlapped with the weight-table computation below ----
    if (lvl == 2) {
        const float* gsrc = feat2 + ((size_t)b * C + c0) * (size_t)LVL2_HW;
        const uint32_t lds0 = (uint32_t)(size_t)(&s_planes[0]);
        const int total = CT * LVL2_HW * 4;           // 51200 bytes
        // stride 4096B: last pass covers tid<128 -> whole waves, no partial EXEC
        for (int u = tid * 16; u < total; u += BLK * 16) {
            uint32_t la = lds0 + (uint32_t)u;
            asm volatile("global_load_async_to_lds_b128 %0, %1, %2"
                         :: "v"(la), "v"(u), "s"(gsrc)
                         : "memory");
        }
    }

    // ---- per-roi bilinear sample table (196 entries, one pass) ----
    const float x1 = boxes[roi * 4 + 0] * scale;
    const float y1 = boxes[roi * 4 + 1] * scale;
    const float x2 = boxes[roi * 4 + 2] * scale;
    const float y2 = boxes[roi * 4 + 3] * scale;
    const float bw = fmaxf(x2 - x1, 1.f) * (1.f / 7.f);
    const float bh = fmaxf(y2 - y1, 1.f) * (1.f / 7.f);

    if (tid < NSAMP) {
        const int e   = tid;
        const int ix  = e & 1;
        const int iy  = (e >> 1) & 1;
        const int bin = e >> 2;
        const int ph  = bin / 7;
        const int pw  = bin - ph * 7;
        float yy = y1 + bh * ((float)ph + ((float)iy + 0.5f) * 0.5f);
        float xx = x1 + bw * ((float)pw + ((float)ix + 0.5f) * 0.5f);
        const bool valid = (yy > -1.f) && (yy < (float)H) &&
                           (xx > -1.f) && (xx < (float)W);
        yy = fminf(fmaxf(yy, 0.f), (float)(H - 1));
        xx = fminf(fmaxf(xx, 0.f), (float)(W - 1));
        const int   yl = (int)floorf(yy);
        const int   xl = (int)floorf(xx);
        const int   yh = min(yl + 1, H - 1);
        const int   xh = min(xl + 1, W - 1);
        const float ly = yy - (float)yl;
        const float lx = xx - (float)xl;
        const float hy = 1.f - ly;
        const float hx = 1.f - lx;
        const float q  = valid ? 0.25f : 0.f;   // fold the 4-sample mean
        s_w[e]   = make_float4(hy * hx * q, hy * lx * q, ly * hx * q, ly * lx * q);
        s_off[e] = make_int4(yl * W + xl, yl * W + xh, yh * W + xl, yh * W + xh);
    }

    if (lvl == 2)
        asm volatile("s_wait_asynccnt 0x0" ::: "memory");
    __syncthreads();

    // ---- main accumulation: 32 channels x 49 bins = 1568 = 6*256 + 32 ----
    if (lvl == 2) {
#pragma unroll
        for (int i = 0; i < 6; ++i)
            accum_item(tid + i * BLK, roi, c0, C, LVL2_HW,
                       &s_planes[0], s_off, s_w, out);
        if (tid < CT * NBIN - 6 * BLK)
            accum_item(tid + 6 * BLK, roi, c0, C, LVL2_HW,
                       &s_planes[0], s_off, s_w, out);
    } else {
        const float* __restrict__ base = feat + ((size_t)b * C + c0) * (size_t)HW;
#pragma unroll
        for (int i = 0; i < 6; ++i)
            accum_item(tid + i * BLK, roi, c0, C, HW, base, s_off, s_w, out);
        if (tid < CT * NBIN - 6 * BLK)
            accum_item(tid + 6 * BLK, roi, c0, C, HW, base, s_off, s_w, out);
    }
}

extern "C" void kernel_launch(void* const* d_in, const int* in_sizes, int n_in,
                              void* d_out, int out_size, void* d_ws, size_t ws_size,
                              hipStream_t stream) {
    const float* feat0 = (const float*)d_in[0];
    const float* feat1 = (const float*)d_in[1];
    const float* feat2 = (const float*)d_in[2];
    const float* boxes = (const float*)d_in[3];
    const int*   bids  = (const int*)d_in[4];
    const int*   fidx  = (const int*)d_in[5];
    float*       outp  = (float*)d_out;

    const int N = in_sizes[3] / 4;                 // 1024 rois
    const int C = in_sizes[0] / (2 * 80 * 80);     // B=2 -> C=256
    const int ntiles = C / CT;

    if (ntiles == 8) {
        roi_align_cdna5_kernel<8><<<dim3(N * 8), dim3(BLK), 0, stream>>>(
            feat0, feat1, feat2, boxes, bids, fidx, outp, C);
    } else if (ntiles == 4) {
        roi_align_cdna5_kernel<4><<<dim3(N * 4), dim3(BLK), 0, stream>>>(
            feat0, feat1, feat2, boxes, bids, fidx, outp, C);
    } else {
        roi_align_cdna5_kernel<1><<<dim3(N * ntiles), dim3(BLK), 0, stream>>>(
            feat0, feat1, feat2, boxes, bids, fidx, outp, C);
    }
}